// DisLoss_49340584297211
// MI455X (gfx1250) — compile-verified
//
#include <hip/hip_runtime.h>
#include <math.h>

typedef __attribute__((ext_vector_type(2))) float v2f;
typedef __attribute__((ext_vector_type(8))) float v8f;

static constexpr int H = 512;
static constexpr int W = 512;
static constexpr int HW = H * W;
static constexpr int RG_PER_B = 8;           // row-groups per batch in pass 2
static constexpr int ROWS_PER_RG = H / RG_PER_B;  // 64

// ---------------------------------------------------------------------------
// Exact wave32 sum reduction built from two V_WMMA_F32_16X16X4_F32 ops.
// Step 1: A[lane%16][lane<16?0:2] = p, B = ones  -> D[m][n] = p_m + p_{m+16}
// Step 2: per-lane sum of 8 D VGPRs -> half-uniform partials S_lo / S_hi
// Step 3: A = ones, B holds S        -> every D element = S_lo + S_hi = total
// EXEC must be all ones at call sites (it is: no divergence there).
// ---------------------------------------------------------------------------
__device__ __forceinline__ float wave_sum_wmma(float p) {
    v2f a;    a[0] = p;    a[1] = 0.0f;
    v2f ones; ones[0] = 1.0f; ones[1] = 1.0f;
    v8f c = {};
    v8f d = __builtin_amdgcn_wmma_f32_16x16x4_f32(
        false, a, false, ones, (short)0, c, false, false);
    float s = 0.0f;
#pragma unroll
    for (int i = 0; i < 8; ++i) s += d[i];
    v2f b2; b2[0] = s; b2[1] = 0.0f;
    v8f d2 = __builtin_amdgcn_wmma_f32_16x16x4_f32(
        false, ones, false, b2, (short)0, c, false, false);
    return d2[0];
}

// ---------------------------------------------------------------------------
// Pass 1: per-batch argmax (first occurrence on ties, matching jnp.argmax).
// One 1024-thread block per batch; 64 float4 loads per thread, coalesced.
// ---------------------------------------------------------------------------
__global__ __launch_bounds__(1024)
void disloss_argmax_kernel(const float* __restrict__ x, int* __restrict__ widx) {
    const int b = blockIdx.x;
    const float4* xb = (const float4*)(x + (size_t)b * HW);
    const int tid = threadIdx.x;

    float bv = -INFINITY;
    int   bi = 0x7fffffff;

#pragma unroll 4
    for (int k = 0; k < HW / 4 / 1024; ++k) {   // 64 iterations
        const int q = tid + (k << 10);
        const float4 v = xb[q];
        const int base = q << 2;
        if (v.x > bv || (v.x == bv && base     < bi)) { bv = v.x; bi = base;     }
        if (v.y > bv || (v.y == bv && base + 1 < bi)) { bv = v.y; bi = base + 1; }
        if (v.z > bv || (v.z == bv && base + 2 < bi)) { bv = v.z; bi = base + 2; }
        if (v.w > bv || (v.w == bv && base + 3 < bi)) { bv = v.w; bi = base + 3; }
    }

    // wave32 reduce (value, lowest index wins on tie)
#pragma unroll
    for (int off = 16; off > 0; off >>= 1) {
        const float ov = __shfl_xor(bv, off, 32);
        const int   oi = __shfl_xor(bi, off, 32);
        if (ov > bv || (ov == bv && oi < bi)) { bv = ov; bi = oi; }
    }

    __shared__ float sv[32];
    __shared__ int   si[32];
    const int wave = tid >> 5;
    const int lane = tid & 31;
    if (lane == 0) { sv[wave] = bv; si[wave] = bi; }
    __syncthreads();

    if (wave == 0) {
        bv = sv[lane];
        bi = si[lane];
#pragma unroll
        for (int off = 16; off > 0; off >>= 1) {
            const float ov = __shfl_xor(bv, off, 32);
            const int   oi = __shfl_xor(bi, off, 32);
            if (ov > bv || (ov == bv && oi < bi)) { bv = ov; bi = oi; }
        }
        if (lane == 0) widx[b] = bi;
    }
}

// ---------------------------------------------------------------------------
// Pass 2: weighted sum. grid = B * 8 blocks of 256 threads; block (b, rg)
// covers 64 rows x 512 cols. Likely L2-resident after pass 1 (128MB < 192MB).
// Deterministic: per-block partial written to fixed slot, no atomics.
// ---------------------------------------------------------------------------
__global__ __launch_bounds__(256)
void disloss_wsum_kernel(const float* __restrict__ x,
                         const int* __restrict__ widx,
                         float* __restrict__ part) {
    const int blk = blockIdx.x;
    const int b   = blk >> 3;
    const int rg  = blk & (RG_PER_B - 1);

    const int   idx = widx[b];
    const float cy  = (float)(idx >> 9);     // idx / 512
    const float cx  = (float)(idx & 511);    // idx % 512

    const float4* xb = (const float4*)(x + (size_t)b * HW);
    const int tid = threadIdx.x;
    const int c4  = tid & 127;               // float4 column (512/4 = 128)
    const int r0  = rg * ROWS_PER_RG + (tid >> 7);

    const float fx0 = (float)(c4 << 2) - cx;
    const float fx1 = fx0 + 1.0f;
    const float fx2 = fx0 + 2.0f;
    const float fx3 = fx0 + 3.0f;

    float acc = 0.0f;
#pragma unroll 4
    for (int r = r0; r < (rg + 1) * ROWS_PER_RG; r += 2) {
        const float dy  = (float)r - cy;
        const float dy2 = dy * dy;
        const float4 v  = xb[(r << 7) + c4];
        const float d0 = __builtin_amdgcn_sqrtf(__builtin_fmaf(fx0, fx0, dy2));
        const float d1 = __builtin_amdgcn_sqrtf(__builtin_fmaf(fx1, fx1, dy2));
        const float d2 = __builtin_amdgcn_sqrtf(__builtin_fmaf(fx2, fx2, dy2));
        const float d3 = __builtin_amdgcn_sqrtf(__builtin_fmaf(fx3, fx3, dy2));
        acc = __builtin_fmaf(v.x, d0, acc);
        acc = __builtin_fmaf(v.y, d1, acc);
        acc = __builtin_fmaf(v.z, d2, acc);
        acc = __builtin_fmaf(v.w, d3, acc);
    }

    const float wsum = wave_sum_wmma(acc);   // v_wmma_f32_16x16x4_f32 x2

    __shared__ float sw[8];
    const int wave = tid >> 5;
    const int lane = tid & 31;
    if (lane == 0) sw[wave] = wsum;
    __syncthreads();
    if (tid == 0) {
        float t = 0.0f;
#pragma unroll
        for (int i = 0; i < 8; ++i) t += sw[i];
        part[blk] = t;
    }
}

// ---------------------------------------------------------------------------
// Pass 3: single-block deterministic final reduce over B*8 partials.
// ---------------------------------------------------------------------------
__global__ __launch_bounds__(1024)
void disloss_final_kernel(const float* __restrict__ part, int n, float invB,
                          float* __restrict__ out) {
    float acc = 0.0f;
    for (int i = threadIdx.x; i < n; i += 1024) acc += part[i];

    const float wsum = wave_sum_wmma(acc);

    __shared__ float sw[32];
    const int wave = threadIdx.x >> 5;
    const int lane = threadIdx.x & 31;
    if (lane == 0) sw[wave] = wsum;
    __syncthreads();
    if (threadIdx.x == 0) {
        float t = 0.0f;
#pragma unroll
        for (int i = 0; i < 32; ++i) t += sw[i];
        out[0] = t * invB;
    }
}

extern "C" void kernel_launch(void* const* d_in, const int* in_sizes, int n_in,
                              void* d_out, int out_size, void* d_ws, size_t ws_size,
                              hipStream_t stream) {
    const float* x = (const float*)d_in[0];
    const int B = in_sizes[0] / HW;          // 128 for the reference shape

    int*   widx = (int*)d_ws;                         // B ints
    float* part = (float*)((char*)d_ws + 4096);       // B*8 floats

    disloss_argmax_kernel<<<B, 1024, 0, stream>>>(x, widx);
    disloss_wsum_kernel<<<B * RG_PER_B, 256, 0, stream>>>(x, widx, part);
    disloss_final_kernel<<<1, 1024, 0, stream>>>(part, B * RG_PER_B,
                                                 1.0f / (float)B, (float*)d_out);
}